// ConcatNet_5781025980901
// MI455X (gfx1250) — compile-verified
//
#include <hip/hip_runtime.h>
#include <cstddef>

// ---------------------------------------------------------------------------
// CDNA5 fp32 WMMA helpers (V_WMMA_F32_16X16X4_F32, wave32)
//   A (16x4 f32):  A[m][k], m = lane&15, k = 2*(lane>>4) + v   (v=0..1)
//   B (4x16 f32):  B[k][n], n = lane&15, k = 2*(lane>>4) + v
//   C/D (16x16):   D[m][n], n = lane&15, m = 8*(lane>>4) + v   (v=0..7)
// ---------------------------------------------------------------------------
typedef float v2f __attribute__((ext_vector_type(2)));
typedef float v8f __attribute__((ext_vector_type(8)));

// Dual row-tile accumulate: two 16x16 output tiles (rows 0-15 and 16-31 of the
// LDS A panel) sharing every B fragment -> 1 global B pair per 2 WMMAs.
__device__ __forceinline__ void wmma_tile_dual(const float* __restrict__ A, int lda,
                                               const float* __restrict__ B, int ldb,
                                               int K, v8f& acc0, v8f& acc1) {
  const int lane = threadIdx.x & 31;
  const int r  = lane & 15;          // M for A, N for B
  const int kk = (lane >> 4) << 1;   // 0 or 2
  const float* __restrict__ A1 = A + 16 * lda;
  for (int k = 0; k < K; k += 4) {
    v2f b, a0, a1;
    b.x  = B[(size_t)(k + kk) * ldb + r];
    b.y  = B[(size_t)(k + kk + 1) * ldb + r];
    a0.x = A [r * lda + (k + kk)];
    a0.y = A [r * lda + (k + kk + 1)];
    a1.x = A1[r * lda + (k + kk)];
    a1.y = A1[r * lda + (k + kk + 1)];
    acc0 = __builtin_amdgcn_wmma_f32_16x16x4_f32(false, a0, false, b,
                                                 (short)0, acc0, false, false);
    acc1 = __builtin_amdgcn_wmma_f32_16x16x4_f32(false, a1, false, b,
                                                 (short)0, acc1, false, false);
  }
}

// Single row-tile version (final projection).
__device__ __forceinline__ v8f wmma_tile(const float* __restrict__ A, int lda,
                                         const float* __restrict__ B, int ldb,
                                         int K, v8f acc) {
  const int lane = threadIdx.x & 31;
  const int r  = lane & 15;
  const int kk = (lane >> 4) << 1;
  for (int k = 0; k < K; k += 4) {
    v2f a, b;
    a.x = A[r * lda + (k + kk)];
    a.y = A[r * lda + (k + kk + 1)];
    b.x = B[(size_t)(k + kk) * ldb + r];
    b.y = B[(size_t)(k + kk + 1) * ldb + r];
    acc = __builtin_amdgcn_wmma_f32_16x16x4_f32(false, a, false, b,
                                                (short)0, acc, false, false);
  }
  return acc;
}

// Store a 16x16 D tile into LDS with bias (+ optional ReLU). bias indexed by n.
__device__ __forceinline__ void tile_store_lds(float* __restrict__ O, int ldo,
                                               v8f d, const float* __restrict__ bias,
                                               bool relu) {
  const int lane = threadIdx.x & 31;
  const int n  = lane & 15;
  const int m0 = (lane >> 4) * 8;
#pragma unroll
  for (int v = 0; v < 8; ++v) {
    float t = d[v] + bias[n];
    if (relu) t = fmaxf(t, 0.0f);
    O[(m0 + v) * ldo + n] = t;
  }
}

// LDS strides with (stride mod 8) == 4 -> conflict-free WMMA A-column reads.
#define LDS_X   132   // 128-wide panels
#define LDS_H   260   // 256-wide panels
#define LDS_M   324   // 320-wide panels

// ---------------------------------------------------------------------------
// Node MLP: xn = relu(x @ W1[128,256] + b1) @ W2[256,128] + b2
// 32 rows per 256-thread block (8 waves), dual row tiles, tail-guarded.
// ---------------------------------------------------------------------------
__global__ void __launch_bounds__(256)
node_mlp_kernel(const float* __restrict__ xin,
                const float* __restrict__ W1, const float* __restrict__ b1,
                const float* __restrict__ W2, const float* __restrict__ b2,
                float* __restrict__ xn, int N) {
  __shared__ float sx[32 * LDS_X];
  __shared__ float sh[32 * LDS_H];
  const int tid  = threadIdx.x;
  const int wave = tid >> 5;
  const int r0   = blockIdx.x * 32;

  for (int i = tid; i < 32 * 128; i += 256) {
    int r = i >> 7, c = i & 127;
    int gr = r0 + r;
    sx[r * LDS_X + c] = (gr < N) ? xin[(size_t)gr * 128 + c] : 0.0f;
  }
  __syncthreads();

  // stage 1: hidden [32,256] -> 16 col tiles x 2 row tiles, 2 col tiles/wave
  for (int ct = wave; ct < 16; ct += 8) {
    v8f acc0 = {}, acc1 = {};
    wmma_tile_dual(sx, LDS_X, W1 + ct * 16, 256, 128, acc0, acc1);
    tile_store_lds(sh + ct * 16,              LDS_H, acc0, b1 + ct * 16, true);
    tile_store_lds(sh + 16 * LDS_H + ct * 16, LDS_H, acc1, b1 + ct * 16, true);
  }
  __syncthreads();

  // stage 2: out [32,128] -> 8 col tiles, 1/wave, dual row tiles
  {
    const int ct = wave;
    v8f acc0 = {}, acc1 = {};
    wmma_tile_dual(sh, LDS_H, W2 + ct * 16, 128, 256, acc0, acc1);
    const int lane = tid & 31;
    const int n  = lane & 15;
    const int m0 = (lane >> 4) * 8;
#pragma unroll
    for (int v = 0; v < 8; ++v) {
      int gr0 = r0 + m0 + v;
      int gr1 = r0 + 16 + m0 + v;
      float bv = b2[ct * 16 + n];
      if (gr0 < N) xn[(size_t)gr0 * 128 + ct * 16 + n] = acc0[v] + bv;
      if (gr1 < N) xn[(size_t)gr1 * 128 + ct * 16 + n] = acc1[v] + bv;
    }
  }
}

// ---------------------------------------------------------------------------
// Fused per-edge pipeline, 32 edges per 256-thread block:
//   e  = relu(edge_attr @ We1[64,128] + be1) @ We2[128,64] + be2
//   mi = [xn[dst] | xn[src] | e]                       (32 x 320)
//   m  = relu(mi @ Wm1[320,128] + bm1) @ Wm2[128,128] + bm2
//   atomic scatter m into agg[dst]
// Raw edge_attr is staged in the concat buffer's e-columns (256..319) and
// overwritten in place by the edge-MLP output (barrier-separated).
// ---------------------------------------------------------------------------
__global__ void __launch_bounds__(256)
message_kernel(const float* __restrict__ xn, const float* __restrict__ ea,
               const int* __restrict__ srcI, const int* __restrict__ dstI,
               const float* __restrict__ We1, const float* __restrict__ be1,
               const float* __restrict__ We2, const float* __restrict__ be2,
               const float* __restrict__ Wm1, const float* __restrict__ bm1,
               const float* __restrict__ Wm2, const float* __restrict__ bm2,
               float* __restrict__ agg) {
  __shared__ float smin[32 * LDS_M];  // concat input [32,320]
  __shared__ float sh2[32 * LDS_X];   // hidden buffer [32,128] (reused)
  __shared__ int   sdst[32], ssrc[32];

  const int tid  = threadIdx.x;
  const int wave = tid >> 5;
  const int lane = tid & 31;
  const int e0   = blockIdx.x * 32;

  if (tid < 32)       sdst[tid]      = dstI[e0 + tid];
  else if (tid < 64)  ssrc[tid - 32] = srcI[e0 + tid - 32];
  // raw edge_attr -> smin cols 256..319
  for (int i = tid; i < 32 * 64; i += 256) {
    int r = i >> 6, c = i & 63;
    smin[r * LDS_M + 256 + c] = ea[(size_t)(e0 + r) * 64 + c];
  }
  __syncthreads();

  // gather xn[dst] -> cols 0..127, xn[src] -> cols 128..255
  for (int i = tid; i < 32 * 128; i += 256) {
    int r = i >> 7, c = i & 127;
    smin[r * LDS_M + c]       = xn[(size_t)sdst[r] * 128 + c];
    smin[r * LDS_M + 128 + c] = xn[(size_t)ssrc[r] * 128 + c];
  }
  // edge MLP stage 1: eh = relu(edge_attr @ We1 + be1)  [32,128], K=64
  {
    const int ct = wave;
    v8f acc0 = {}, acc1 = {};
    wmma_tile_dual(smin + 256, LDS_M, We1 + ct * 16, 128, 64, acc0, acc1);
    tile_store_lds(sh2 + ct * 16,              LDS_X, acc0, be1 + ct * 16, true);
    tile_store_lds(sh2 + 16 * LDS_X + ct * 16, LDS_X, acc1, be1 + ct * 16, true);
  }
  __syncthreads();   // sh2 ready; all stage-1 reads of edge_attr columns done

  // edge MLP stage 2: e = eh @ We2 + be2 [32,64] -> overwrite smin cols 256+
  if (wave < 4) {
    const int ct = wave;
    v8f acc0 = {}, acc1 = {};
    wmma_tile_dual(sh2, LDS_X, We2 + ct * 16, 64, 128, acc0, acc1);
    tile_store_lds(smin + 256 + ct * 16,              LDS_M, acc0, be2 + ct * 16, false);
    tile_store_lds(smin + 16 * LDS_M + 256 + ct * 16, LDS_M, acc1, be2 + ct * 16, false);
  }
  __syncthreads();   // smin fully assembled; sh2 reads done

  // message MLP stage 1: h = relu(smin @ Wm1 + bm1)  [32,128], K=320
  {
    const int ct = wave;
    v8f acc0 = {}, acc1 = {};
    wmma_tile_dual(smin, LDS_M, Wm1 + ct * 16, 128, 320, acc0, acc1);
    tile_store_lds(sh2 + ct * 16,              LDS_X, acc0, bm1 + ct * 16, true);
    tile_store_lds(sh2 + 16 * LDS_X + ct * 16, LDS_X, acc1, bm1 + ct * 16, true);
  }
  __syncthreads();

  // message MLP stage 2 + scatter-add: m = h @ Wm2 + bm2
  {
    const int ct = wave;
    v8f acc0 = {}, acc1 = {};
    wmma_tile_dual(sh2, LDS_X, Wm2 + ct * 16, 128, 128, acc0, acc1);
    const int n  = lane & 15;
    const int m0 = (lane >> 4) * 8;
    const float bv = bm2[ct * 16 + n];
#pragma unroll
    for (int v = 0; v < 8; ++v) {
      atomicAdd(&agg[(size_t)sdst[m0 + v]      * 128 + ct * 16 + n], acc0[v] + bv);
      atomicAdd(&agg[(size_t)sdst[16 + m0 + v] * 128 + ct * 16 + n], acc1[v] + bv);
    }
  }
}

// ---------------------------------------------------------------------------
// Scatter-mean divide + LayerNorm + ReLU. One node per 128-thread block.
// ---------------------------------------------------------------------------
__global__ void __launch_bounds__(128)
norm_kernel(const float* __restrict__ agg, const float* __restrict__ deg,
            const float* __restrict__ lnw, const float* __restrict__ lnb,
            float* __restrict__ xout) {
  __shared__ float red[8];
  const int nid = blockIdx.x;
  const int c   = threadIdx.x;
  const float denom = fmaxf(deg[nid], 1.0f);
  const float val = agg[(size_t)nid * 128 + c] / denom;

  float s = val;
#pragma unroll
  for (int o = 16; o > 0; o >>= 1) s += __shfl_xor(s, o, 32);
  if ((c & 31) == 0) red[c >> 5] = s;
  __syncthreads();
  if (c < 32) {
    float t = (c < 4) ? red[c] : 0.0f;
    t += __shfl_xor(t, 2, 32);
    t += __shfl_xor(t, 1, 32);
    if (c == 0) red[4] = t;
  }
  __syncthreads();
  const float mu = red[4] * (1.0f / 128.0f);
  const float d  = val - mu;
  float sq = d * d;
#pragma unroll
  for (int o = 16; o > 0; o >>= 1) sq += __shfl_xor(sq, o, 32);
  if ((c & 31) == 0) red[c >> 5] = sq;
  __syncthreads();
  if (c < 32) {
    float t = (c < 4) ? red[c] : 0.0f;
    t += __shfl_xor(t, 2, 32);
    t += __shfl_xor(t, 1, 32);
    if (c == 0) red[5] = t;
  }
  __syncthreads();
  const float var = red[5] * (1.0f / 128.0f);
  const float h = d * rsqrtf(var + 1e-5f) * lnw[c] + lnb[c];
  xout[(size_t)nid * 128 + c] = fmaxf(h, 0.0f);
}

// ---------------------------------------------------------------------------
// Final projection: out = x @ Wf[128,64] + bf.  16 rows per 128-thread block.
// ---------------------------------------------------------------------------
__global__ void __launch_bounds__(128)
final_kernel(const float* __restrict__ x, const float* __restrict__ Wf,
             const float* __restrict__ bf, float* __restrict__ out) {
  __shared__ float sx[16 * LDS_X];
  const int tid  = threadIdx.x;
  const int wave = tid >> 5;  // 0..3
  const int r0   = blockIdx.x * 16;
  for (int i = tid; i < 16 * 128; i += 128) {
    int r = i >> 7, c = i & 127;
    sx[r * LDS_X + c] = x[(size_t)(r0 + r) * 128 + c];
  }
  __syncthreads();
  const int ct = wave;
  v8f acc = {};
  acc = wmma_tile(sx, LDS_X, Wf + ct * 16, 64, 128, acc);
  const int lane = tid & 31;
  const int n  = lane & 15;
  const int m0 = (lane >> 4) * 8;
#pragma unroll
  for (int v = 0; v < 8; ++v)
    out[(size_t)(r0 + m0 + v) * 64 + ct * 16 + n] = acc[v] + bf[ct * 16 + n];
}

// ---------------------------------------------------------------------------
// Utilities
// ---------------------------------------------------------------------------
__global__ void zero_kernel(float* __restrict__ p, size_t n) {
  size_t i = (size_t)blockIdx.x * blockDim.x + threadIdx.x;
  if (i < n) p[i] = 0.0f;
}

__global__ void deg_kernel(const int* __restrict__ dstI, float* __restrict__ deg, int E) {
  int i = blockIdx.x * blockDim.x + threadIdx.x;
  if (i < E) atomicAdd(&deg[dstI[i]], 1.0f);
}

// ---------------------------------------------------------------------------
extern "C" void kernel_launch(void* const* d_in, const int* in_sizes, int n_in,
                              void* d_out, int out_size, void* d_ws, size_t ws_size,
                              hipStream_t stream) {
  const float* x    = (const float*)d_in[0];
  const float* ea   = (const float*)d_in[1];
  const int*   ei   = (const int*)d_in[2];
  const float* Wn1  = (const float*)d_in[3];
  const float* bn1  = (const float*)d_in[4];
  const float* Wn2  = (const float*)d_in[5];
  const float* bn2  = (const float*)d_in[6];
  const float* We1  = (const float*)d_in[7];
  const float* be1  = (const float*)d_in[8];
  const float* We2  = (const float*)d_in[9];
  const float* be2  = (const float*)d_in[10];
  const float* Wm1  = (const float*)d_in[11];
  const float* bm1  = (const float*)d_in[12];
  const float* Wm2  = (const float*)d_in[13];
  const float* bm2  = (const float*)d_in[14];
  const float* lnw  = (const float*)d_in[15];
  const float* lnb  = (const float*)d_in[16];
  const float* Wf   = (const float*)d_in[17];
  const float* bf   = (const float*)d_in[18];
  float* out = (float*)d_out;

  const int N = in_sizes[0] / 128;   // 50000
  const int E = in_sizes[1] / 64;    // 800000
  const int L = 3;

  const int* srcI = ei;       // edge_index[0]
  const int* dstI = ei + E;   // edge_index[1]

  float* xn   = (float*)d_ws;
  float* agg  = xn  + (size_t)N * 128;
  float* xcur = agg + (size_t)N * 128;
  float* deg  = xcur + (size_t)N * 128;

  zero_kernel<<<(N + 255) / 256, 256, 0, stream>>>(deg, (size_t)N);
  deg_kernel<<<(E + 255) / 256, 256, 0, stream>>>(dstI, deg, E);

  const float* xsrc = x;
  for (int l = 0; l < L; ++l) {
    node_mlp_kernel<<<(N + 31) / 32, 256, 0, stream>>>(
        xsrc, Wn1 + (size_t)l * 128 * 256, bn1 + (size_t)l * 256,
        Wn2 + (size_t)l * 256 * 128, bn2 + (size_t)l * 128, xn, N);
    zero_kernel<<<(int)(((size_t)N * 128 + 255) / 256), 256, 0, stream>>>(
        agg, (size_t)N * 128);
    message_kernel<<<E / 32, 256, 0, stream>>>(
        xn, ea, srcI, dstI,
        We1 + (size_t)l * 64 * 128,  be1 + (size_t)l * 128,
        We2 + (size_t)l * 128 * 64,  be2 + (size_t)l * 64,
        Wm1 + (size_t)l * 320 * 128, bm1 + (size_t)l * 128,
        Wm2 + (size_t)l * 128 * 128, bm2 + (size_t)l * 128, agg);
    norm_kernel<<<N, 128, 0, stream>>>(agg, deg,
                                       lnw + (size_t)l * 128,
                                       lnb + (size_t)l * 128, xcur);
    xsrc = xcur;
  }
  final_kernel<<<N / 16, 128, 0, stream>>>(xcur, Wf, bf, out);
}